// TransformerUpdate_32186484916932
// MI455X (gfx1250) — compile-verified
//
#include <hip/hip_runtime.h>

typedef __attribute__((ext_vector_type(2))) float v2f;
typedef __attribute__((ext_vector_type(8))) float v8f;

#define FEAT  120
#define NHEAD 4
#define XSTR  121   // padded LDS row stride (floats)

// ---------------------------------------------------------------------------
// One 16x16 output tile of Y[node, out_chan] = sum_k A[node,k] * B[k,out_chan],
// chained over K in steps of 4 via V_WMMA_F32_16X16X4_F32, computing NC
// spherical components at once (B fragment shared across components).
//
// A layout (ISA 7.12.2, 32-bit A 16x4):  lane L, vgpr j -> A[M=L%16, K=2*(L/16)+j]
// B layout (4x16, row striped):          lane L, vgpr j -> B[K=2*(L/16)+j, N=L%16]
// D layout (16x16 f32):                  lane L, vgpr r -> D[M=r+8*(L/16), N=L%16]
// ---------------------------------------------------------------------------
template <int NC>
__device__ __forceinline__ void wmma_block(const float* __restrict__ xs, int off, int tw,
                                           const float* __restrict__ w, float wscale,
                                           int m, int sK_, int sO_, int ncol0,
                                           int lrow, int half, v8f acc[NC]) {
  const int o = ncol0 + lrow;
  const bool ok = (o < m);
#pragma unroll
  for (int k0 = 0; k0 < m; k0 += 4) {
    const int ka = k0 + 2 * half;
    v2f b;
    b.x = ok ? w[ka * sK_ + o * sO_] * wscale : 0.0f;
    b.y = ok ? w[(ka + 1) * sK_ + o * sO_] * wscale : 0.0f;
#pragma unroll
    for (int c = 0; c < NC; ++c) {
      v2f a;
      a.x = xs[lrow * XSTR + off + ka * tw + c];
      a.y = xs[lrow * XSTR + off + (ka + 1) * tw + c];
      acc[c] = __builtin_amdgcn_wmma_f32_16x16x4_f32(false, a, false, b, (short)0,
                                                     acc[c], false, false);
    }
  }
}

__device__ __forceinline__ void nact_store(float* __restrict__ g, float* __restrict__ g2,
                                           float* __restrict__ lds, int n0, int N,
                                           int node, int feat, float val) {
  if (lds) lds[node * XSTR + feat] = val;
  if (g != nullptr && (n0 + node) < N) {
    size_t idx = (size_t)(n0 + node) * FEAT + feat;
    g[idx] = val;
    if (g2) g2[idx] = val;
  }
}

// o3_linear (1/sqrt(m) folded) followed by NormActivation(sigmoid).
__device__ __forceinline__ void linear_norm(const float* __restrict__ xs,
                                            const float* __restrict__ W0,
                                            const float* __restrict__ W1,
                                            const float* __restrict__ W2,
                                            float* __restrict__ g, float* __restrict__ g2,
                                            float* __restrict__ lds,
                                            int n0, int N, int lrow, int half) {
  const float s0 = 0.17677669529663687f;   // 1/sqrt(32)
  const float s1 = 0.25f;                  // 1/sqrt(16)
  const float s2 = 0.35355339059327373f;   // 1/sqrt(8)
  const float eps2 = 1e-10f;               // EPS^2

  // ---- l=0 block: feats [0,32), K=32, N=32 (two 16-col tiles) ----
#pragma unroll
  for (int t = 0; t < 2; ++t) {
    v8f acc[1] = {};
    wmma_block<1>(xs, 0, 1, W0, s0, 32, 32, 1, t * 16, lrow, half, acc);
#pragma unroll
    for (int r = 0; r < 8; ++r) {
      int node = r + 8 * half;
      float y = acc[0][r];
      float nrm = sqrtf(y * y + eps2);
      float val = y / (nrm * (1.0f + __expf(-nrm)));   // sigmoid(n)*y/n
      nact_store(g, g2, lds, n0, N, node, t * 16 + lrow, val);
    }
  }
  // ---- l=1 block: feats [32,80), K=16, N=16, 3 components ----
  {
    v8f acc[3] = {};
    wmma_block<3>(xs, 32, 3, W1, s1, 16, 16, 1, 0, lrow, half, acc);
#pragma unroll
    for (int r = 0; r < 8; ++r) {
      int node = r + 8 * half;
      float n2 = acc[0][r] * acc[0][r] + acc[1][r] * acc[1][r] + acc[2][r] * acc[2][r] + eps2;
      float nrm = sqrtf(n2);
      float sc = 1.0f / (nrm * (1.0f + __expf(-nrm)));
      int fb = 32 + lrow * 3;
#pragma unroll
      for (int c = 0; c < 3; ++c)
        nact_store(g, g2, lds, n0, N, node, fb + c, acc[c][r] * sc);
    }
  }
  // ---- l=2 block: feats [80,120), K=8, N=8 (cols >=8 masked), 5 components ----
  {
    v8f acc[5] = {};
    wmma_block<5>(xs, 80, 5, W2, s2, 8, 8, 1, 0, lrow, half, acc);
    if (lrow < 8) {
#pragma unroll
      for (int r = 0; r < 8; ++r) {
        int node = r + 8 * half;
        float n2 = eps2;
#pragma unroll
        for (int c = 0; c < 5; ++c) n2 += acc[c][r] * acc[c][r];
        float nrm = sqrtf(n2);
        float sc = 1.0f / (nrm * (1.0f + __expf(-nrm)));
        int fb = 80 + lrow * 5;
#pragma unroll
        for (int c = 0; c < 5; ++c)
          nact_store(nullptr, nullptr, lds, n0, N, node, fb + c, acc[c][r] * sc);
#pragma unroll
        for (int c = 0; c < 5; ++c)
          nact_store(g, g2, nullptr, n0, N, node, fb + c, acc[c][r] * sc);
      }
    }
  }
}

// kW[n,h,i,m] = sum_j Wd[i,j,h] * k[n,j,m]  (tp_dot norm constants folded into scale)
// Wd element (i,j,h) at i*(m*H)+j*H+h -> B[k=j,n=i]: strideK=H, strideO=m*H, base=Wd+h.
__device__ __forceinline__ void kw_head(const float* __restrict__ ks,
                                        const float* __restrict__ D0,
                                        const float* __restrict__ D1,
                                        const float* __restrict__ D2,
                                        float dsc0, float dsc1, float dsc2,
                                        float* __restrict__ kwg, int h,
                                        int n0, int N, int lrow, int half) {
#pragma unroll
  for (int t = 0; t < 2; ++t) {
    v8f acc[1] = {};
    wmma_block<1>(ks, 0, 1, D0, dsc0, 32, NHEAD, 32 * NHEAD, t * 16, lrow, half, acc);
#pragma unroll
    for (int r = 0; r < 8; ++r) {
      int node = r + 8 * half;
      if (n0 + node < N)
        kwg[((size_t)(n0 + node) * NHEAD + h) * FEAT + t * 16 + lrow] = acc[0][r];
    }
  }
  {
    v8f acc[3] = {};
    wmma_block<3>(ks, 32, 3, D1, dsc1, 16, NHEAD, 16 * NHEAD, 0, lrow, half, acc);
#pragma unroll
    for (int r = 0; r < 8; ++r) {
      int node = r + 8 * half;
      if (n0 + node < N) {
        int fb = 32 + lrow * 3;
#pragma unroll
        for (int c = 0; c < 3; ++c)
          kwg[((size_t)(n0 + node) * NHEAD + h) * FEAT + fb + c] = acc[c][r];
      }
    }
  }
  {
    v8f acc[5] = {};
    wmma_block<5>(ks, 80, 5, D2, dsc2, 8, NHEAD, 8 * NHEAD, 0, lrow, half, acc);
    if (lrow < 8) {
#pragma unroll
      for (int r = 0; r < 8; ++r) {
        int node = r + 8 * half;
        if (n0 + node < N) {
          int fb = 80 + lrow * 5;
#pragma unroll
          for (int c = 0; c < 5; ++c)
            kwg[((size_t)(n0 + node) * NHEAD + h) * FEAT + fb + c] = acc[c][r];
        }
      }
    }
  }
}

// ---------------------------------------------------------------------------
// Node kernel: q = na(lin(x,Wq)), v = na(lin(x,Wv)) (also init out = v),
//              k = na(lin(x,Wk)) -> LDS -> kW = Wd(h) @ k  per head.
// 128 threads = 4 waves; one wave per 16-node tile.
// ---------------------------------------------------------------------------
__global__ __launch_bounds__(128) void node_kernel(
    const float* __restrict__ x,
    const float* __restrict__ Wq0, const float* __restrict__ Wq1, const float* __restrict__ Wq2,
    const float* __restrict__ Wk0, const float* __restrict__ Wk1, const float* __restrict__ Wk2,
    const float* __restrict__ Wv0, const float* __restrict__ Wv1, const float* __restrict__ Wv2,
    const float* __restrict__ Wd0, const float* __restrict__ Wd1, const float* __restrict__ Wd2,
    float* __restrict__ qg, float* __restrict__ vg, float* __restrict__ kwg,
    float* __restrict__ outInit, int N) {
  __shared__ float sX[4][16 * XSTR];
  __shared__ float sK[4][16 * XSTR];

  const int tid = threadIdx.x;
  const int wave = tid >> 5, lane = tid & 31;
  const int lrow = lane & 15, half = lane >> 4;
  const int n0 = (blockIdx.x * 4 + wave) * 16;
  float* xs = sX[wave];
  float* ks = sK[wave];

  for (int i = lane; i < 16 * FEAT; i += 32) {
    int row = i / FEAT, col = i % FEAT;
    xs[row * XSTR + col] = (n0 + row < N) ? x[(size_t)(n0 + row) * FEAT + col] : 0.0f;
  }
  __syncthreads();

  linear_norm(xs, Wq0, Wq1, Wq2, qg, nullptr, nullptr, n0, N, lrow, half);
  linear_norm(xs, Wv0, Wv1, Wv2, vg, outInit, nullptr, n0, N, lrow, half);
  linear_norm(xs, Wk0, Wk1, Wk2, nullptr, nullptr, ks, n0, N, lrow, half);
  __syncthreads();

  const float d0 = 1.0f / sqrtf(2112.0f);              // 1/sqrt(fan)
  const float d1 = d0 * 0.5773502691896258f;           // * 1/sqrt(3)
  const float d2 = d0 * 0.4472135954999579f;           // * 1/sqrt(5)
#pragma unroll
  for (int h = 0; h < NHEAD; ++h)
    kw_head(ks, Wd0 + h, Wd1 + h, Wd2 + h, d0, d1, d2, kwg, h, n0, N, lrow, half);
}

// ---------------------------------------------------------------------------
// Edge kernel A: logits[e,h] = <q[dst], kW[src,h]>; em[e]=mean_h exp; z[dst]+=em.
// One wave per edge; lanes 0..29 each own a float4 of the 120-dim feature.
// ---------------------------------------------------------------------------
__global__ __launch_bounds__(256) void edge_attn_kernel(
    const float* __restrict__ qg, const float* __restrict__ kwg,
    const int* __restrict__ dst, const int* __restrict__ src,
    float* __restrict__ z, float* __restrict__ em, int E) {
  int e = blockIdx.x * 8 + (threadIdx.x >> 5);
  if (e >= E) return;
  int lane = threadIdx.x & 31;
  int d = dst[e], s = src[e];

  float4 qv = make_float4(0.f, 0.f, 0.f, 0.f);
  if (lane < 30) qv = *reinterpret_cast<const float4*>(qg + (size_t)d * FEAT + 4 * lane);

  float acc[NHEAD];
#pragma unroll
  for (int h = 0; h < NHEAD; ++h) {
    float4 kv = make_float4(0.f, 0.f, 0.f, 0.f);
    if (lane < 30)
      kv = *reinterpret_cast<const float4*>(kwg + ((size_t)s * NHEAD + h) * FEAT + 4 * lane);
    acc[h] = qv.x * kv.x + qv.y * kv.y + qv.z * kv.z + qv.w * kv.w;
  }
#pragma unroll
  for (int off = 16; off > 0; off >>= 1)
#pragma unroll
    for (int h = 0; h < NHEAD; ++h) acc[h] += __shfl_xor(acc[h], off, 32);

  if (lane == 0) {
    float m = 0.25f * (__expf(acc[0]) + __expf(acc[1]) + __expf(acc[2]) + __expf(acc[3]));
    em[e] = m;
    atomicAdd(z + d, m);
  }
}

// ---------------------------------------------------------------------------
// Edge kernel B: out[dst] += sqrt(relu(em/z[dst])) * v[src]
// ---------------------------------------------------------------------------
__global__ __launch_bounds__(256) void edge_scatter_kernel(
    const float* __restrict__ vg, const int* __restrict__ dst, const int* __restrict__ src,
    const float* __restrict__ z, const float* __restrict__ em,
    float* __restrict__ out, int E) {
  int e = blockIdx.x * 8 + (threadIdx.x >> 5);
  if (e >= E) return;
  int lane = threadIdx.x & 31;
  int d = dst[e], s = src[e];
  float zz = z[d];
  float alpha = em[e] / (zz > 0.0f ? zz : 1.0f);
  float w = sqrtf(fmaxf(alpha, 0.0f));
  if (lane < 30) {
    float4 vv = *reinterpret_cast<const float4*>(vg + (size_t)s * FEAT + 4 * lane);
    float* o = out + (size_t)d * FEAT + 4 * lane;
    atomicAdd(o + 0, w * vv.x);
    atomicAdd(o + 1, w * vv.y);
    atomicAdd(o + 2, w * vv.z);
    atomicAdd(o + 3, w * vv.w);
  }
}

extern "C" void kernel_launch(void* const* d_in, const int* in_sizes, int n_in,
                              void* d_out, int out_size, void* d_ws, size_t ws_size,
                              hipStream_t stream) {
  const float* x = (const float*)d_in[0];
  const float* W[12];
  for (int i = 0; i < 12; ++i) W[i] = (const float*)d_in[1 + i];
  const int* edge_dst = (const int*)d_in[13];
  const int* edge_src = (const int*)d_in[14];
  const int N = in_sizes[0] / FEAT;
  const int E = in_sizes[13];

  float* qg = (float*)d_ws;                        // N*120
  float* vg = qg + (size_t)N * FEAT;               // N*120
  float* kwg = vg + (size_t)N * FEAT;              // N*480
  float* z = kwg + (size_t)N * FEAT * NHEAD;       // N
  float* em = z + N;                               // E

  hipMemsetAsync(z, 0, (size_t)N * sizeof(float), stream);

  int nblocks = (N + 63) / 64;
  node_kernel<<<nblocks, 128, 0, stream>>>(x, W[0], W[1], W[2], W[3], W[4], W[5],
                                           W[6], W[7], W[8], W[9], W[10], W[11],
                                           qg, vg, kwg, (float*)d_out, N);

  int eblocks = (E + 7) / 8;
  edge_attn_kernel<<<eblocks, 256, 0, stream>>>(qg, kwg, edge_dst, edge_src, z, em, E);
  edge_scatter_kernel<<<eblocks, 256, 0, stream>>>(vg, edge_dst, edge_src, z, em,
                                                   (float*)d_out, E);
}